// GraphAutoencoder_24653112279423
// MI455X (gfx1250) — compile-verified
//
#include <hip/hip_runtime.h>
#include <math.h>

// ---------------------------------------------------------------------------
// GCN graph autoencoder on MI455X (gfx1250, wave32).
// Scatter/atomic bound -> keep GEMMs in exact f32 via V_WMMA_F32_16X16X4_F32.
// ---------------------------------------------------------------------------

typedef __attribute__((ext_vector_type(2))) float v2f;
typedef __attribute__((ext_vector_type(8))) float v8f;

#define IN_CH  256
#define HID_CH 128
#define OUT_CH 64

// ---------------------------------------------------------------------------
// Dense GEMM: Out[M,NCOLS] = A[M,K] @ W[K,NCOLS], one wave per 16-row stripe,
// wave owns all NCOLS output columns (NCOLS/16 accumulators of v8f).
// Fragment layout per CDNA5 ISA 7.12.2 (f32 16x16x4):
//   A: lane l(0..15)=row M=l, VGPR0={K=2g, K=2g+1} where g=lane>>4
//   B: symmetric, N across lanes, K=2g(+1) per VGPR pair
//   C/D: VGPR r -> M = r + 8*g, N = lane&15
// ---------------------------------------------------------------------------
template <int K, int NCOLS>
__global__ __launch_bounds__(32) void gemm_wmma_f32(
    const float* __restrict__ A, const float* __restrict__ W,
    float* __restrict__ Out, int nRows) {
  constexpr int NT = NCOLS / 16;
  const int mbase = blockIdx.x * 16;
  const int lane  = threadIdx.x & 31;
  const int g     = lane >> 4;   // half-wave id
  const int l     = lane & 15;

  v8f acc[NT] = {};

  // clamp load row for ragged tails (stores are guarded below)
  int arow_idx = mbase + l;
  if (arow_idx >= nRows) arow_idx = nRows - 1;
  const float* __restrict__ arow = A + (size_t)arow_idx * K + 2 * g;

#pragma unroll 4
  for (int k = 0; k < K; k += 4) {
    v2f a;
    a.x = arow[k];
    a.y = arow[k + 1];
    const float* __restrict__ wk = W + (size_t)(k + 2 * g) * NCOLS + l;
#pragma unroll
    for (int t = 0; t < NT; ++t) {
      v2f b;
      b.x = wk[t * 16];
      b.y = wk[NCOLS + t * 16];
      acc[t] = __builtin_amdgcn_wmma_f32_16x16x4_f32(
          /*neg_a=*/false, a, /*neg_b=*/false, b,
          /*c_mod=*/(short)0, acc[t], /*reuse_a=*/false, /*reuse_b=*/false);
    }
  }

  // store: acc[t] VGPR r holds M = r + 8*g, N = t*16 + l
#pragma unroll
  for (int t = 0; t < NT; ++t) {
#pragma unroll
    for (int r = 0; r < 8; ++r) {
      const int m = mbase + r + 8 * g;
      if (m < nRows) Out[(size_t)m * NCOLS + t * 16 + l] = acc[t][r];
    }
  }
}

// ---------------------------------------------------------------------------
// Degree accumulation (u32 atomics, deterministic). Self loop added later.
// ---------------------------------------------------------------------------
__global__ __launch_bounds__(256) void deg_kernel(
    const int* __restrict__ dst, unsigned* __restrict__ deg, int nE) {
  const int e = blockIdx.x * blockDim.x + threadIdx.x;
  if (e < nE) atomicAdd(&deg[dst[e]], 1u);
}

__global__ __launch_bounds__(256) void dinv_kernel(
    const unsigned* __restrict__ deg, float* __restrict__ dinv, int n) {
  const int i = blockIdx.x * blockDim.x + threadIdx.x;
  if (i < n) dinv[i] = rsqrtf((float)deg[i] + 1.0f);  // +1 = self loop
}

// ---------------------------------------------------------------------------
// Edge scatter: agg[dst] += dinv[src]*dinv[dst] * H[src]. Warp-per-edge,
// C/32 contiguous channels per lane, HW f32 atomics at agent scope.
// ---------------------------------------------------------------------------
template <int C>
__global__ __launch_bounds__(256) void scatter_edges(
    const int* __restrict__ src, const int* __restrict__ dst,
    const float* __restrict__ dinv, const float* __restrict__ hin,
    float* __restrict__ agg, int nE) {
  const int wid  = (int)((blockIdx.x * (unsigned)blockDim.x + threadIdx.x) >> 5);
  const int lane = threadIdx.x & 31;
  if (wid >= nE) return;
  const int s = src[wid];
  const int d = dst[wid];
  const float nrm = dinv[s] * dinv[d];
  constexpr int PER = C / 32;
  const float* __restrict__ hp = hin + (size_t)s * C + lane * PER;
  float* __restrict__ ap       = agg + (size_t)d * C + lane * PER;
#pragma unroll
  for (int j = 0; j < PER; ++j) {
    __hip_atomic_fetch_add(&ap[j], nrm * hp[j], __ATOMIC_RELAXED,
                           __HIP_MEMORY_SCOPE_AGENT);
  }
}

// ---------------------------------------------------------------------------
// Finish layer: agg[i,c] = act(agg[i,c] + dinv[i]^2 * lin[i,c] + bias[c])
// (self-loop message folded in; relu for layer 1 only). In-place on agg.
// ---------------------------------------------------------------------------
template <int C>
__global__ __launch_bounds__(256) void finish_layer(
    float* __restrict__ agg, const float* __restrict__ lin,
    const float* __restrict__ dinv, const float* __restrict__ bias,
    int n, int doRelu) {
  const int t = blockIdx.x * blockDim.x + threadIdx.x;
  if (t >= n * C) return;
  const int i = t / C;
  const int c = t % C;
  const float di = dinv[i];
  float v = agg[t] + di * di * lin[t] + bias[c];
  if (doRelu) v = fmaxf(v, 0.0f);
  agg[t] = v;
}

// ---------------------------------------------------------------------------
// Decoder: out[e] = sigmoid( dot64(z[src[e]], z[dst[e]]) ). Warp-per-edge,
// 2 channels/lane, wave32 xor-shuffle reduction.
// ---------------------------------------------------------------------------
__global__ __launch_bounds__(256) void decode_kernel(
    const int* __restrict__ src, const int* __restrict__ dst,
    const float* __restrict__ z, float* __restrict__ out, int nE) {
  const int wid  = (int)((blockIdx.x * (unsigned)blockDim.x + threadIdx.x) >> 5);
  const int lane = threadIdx.x & 31;
  if (wid >= nE) return;
  const int s = src[wid];
  const int d = dst[wid];
  const float* __restrict__ zs = z + (size_t)s * OUT_CH + lane * 2;
  const float* __restrict__ zd = z + (size_t)d * OUT_CH + lane * 2;
  float p = zs[0] * zd[0] + zs[1] * zd[1];
#pragma unroll
  for (int off = 16; off > 0; off >>= 1) p += __shfl_xor(p, off, 32);
  if (lane == 0) out[wid] = 1.0f / (1.0f + __expf(-p));
}

// ---------------------------------------------------------------------------
extern "C" void kernel_launch(void* const* d_in, const int* in_sizes, int n_in,
                              void* d_out, int out_size, void* d_ws, size_t ws_size,
                              hipStream_t stream) {
  const float* x   = (const float*)d_in[0];
  const int*   ei  = (const int*)d_in[1];   // [2, E] flat
  const float* W1  = (const float*)d_in[2];
  const float* b1  = (const float*)d_in[3];
  const float* W2  = (const float*)d_in[4];
  const float* b2  = (const float*)d_in[5];

  const int N = in_sizes[0] / IN_CH;
  const int E = in_sizes[1] / 2;
  const int* esrc = ei;        // edge_index[0]
  const int* edst = ei + E;    // edge_index[1]

  // --- workspace layout (all f32 unless noted) ---
  char* w = (char*)d_ws;
  unsigned* deg  = (unsigned*)w; w += (size_t)N * 4;
  float*    dinv = (float*)w;    w += (size_t)N * 4;
  float*    hlin = (float*)w;    w += (size_t)N * HID_CH * 4;   // X @ W1
  float*    hagg = (float*)w;    w += (size_t)N * HID_CH * 4;   // agg1 / h (in place)
  float*    zlin = (float*)w;    w += (size_t)N * OUT_CH * 4;   // h @ W2
  float*    zagg = (float*)w;    w += (size_t)N * OUT_CH * 4;   // agg2 / z (in place)

  hipMemsetAsync(deg,  0, (size_t)N * 4, stream);
  hipMemsetAsync(hagg, 0, (size_t)N * HID_CH * 4, stream);
  hipMemsetAsync(zagg, 0, (size_t)N * OUT_CH * 4, stream);

  const int TPB = 256;
  const int nWarpBlocks = (int)(((size_t)E * 32 + TPB - 1) / TPB);
  const int mTiles = (N + 15) / 16;

  // normalization
  deg_kernel<<<(E + TPB - 1) / TPB, TPB, 0, stream>>>(edst, deg, E);
  dinv_kernel<<<(N + TPB - 1) / TPB, TPB, 0, stream>>>(deg, dinv, N);

  // layer 1: linear -> propagate -> +b1 -> relu
  gemm_wmma_f32<IN_CH, HID_CH><<<mTiles, 32, 0, stream>>>(x, W1, hlin, N);
  scatter_edges<HID_CH><<<nWarpBlocks, TPB, 0, stream>>>(esrc, edst, dinv, hlin, hagg, E);
  finish_layer<HID_CH><<<((size_t)N * HID_CH + TPB - 1) / TPB, TPB, 0, stream>>>(
      hagg, hlin, dinv, b1, N, 1);

  // layer 2: linear -> propagate -> +b2
  gemm_wmma_f32<HID_CH, OUT_CH><<<mTiles, 32, 0, stream>>>(hagg, W2, zlin, N);
  scatter_edges<OUT_CH><<<nWarpBlocks, TPB, 0, stream>>>(esrc, edst, dinv, zlin, zagg, E);
  finish_layer<OUT_CH><<<((size_t)N * OUT_CH + TPB - 1) / TPB, TPB, 0, stream>>>(
      zagg, zlin, dinv, b2, N, 0);

  // inner-product decoder + sigmoid
  decode_kernel<<<nWarpBlocks, TPB, 0, stream>>>(esrc, edst, zagg, (float*)d_out, E);
}